// InvariantMultiACE_73675868995702
// MI455X (gfx1250) — compile-verified
//
#include <hip/hip_runtime.h>
#include <hip/hip_bf16.h>
#include <math.h>

typedef float v2f __attribute__((ext_vector_type(2)));
typedef float v8f __attribute__((ext_vector_type(8)));

namespace {
constexpr int NN    = 32768;
constexpr int NE    = 524288;
constexpr int NELEM = 10;
constexpr int KC    = 32;
constexpr int NSHC  = 16;
constexpr int NB    = 8;
constexpr int NGR   = 16;
constexpr float RMAX    = 5.0f;
constexpr float INV_AVG = 1.0f / 16.0f;
constexpr float SQ2R    = 0.6324555320336759f;  // sqrt(2/RMAX)
constexpr float PI_F    = 3.14159265358979323846f;

// workspace layout (floats)
constexpr size_t SZ_A    = (size_t)NN * KC * NSHC;   // 16,777,216
constexpr size_t OFF_A   = 0;
constexpr size_t OFF_GA  = OFF_A + SZ_A;
constexpr size_t OFF_Y   = OFF_GA + SZ_A;            // NN*KC
constexpr size_t OFF_VM  = OFF_Y + (size_t)NN * KC;  // 4096
constexpr size_t OFF_U2S = OFF_VM + 4096;            // 256
constexpr size_t OFF_WV  = OFF_U2S + 256;            // 96
}

__device__ inline v8f wmma4(v2f a, v2f b, v8f c) {
  return __builtin_amdgcn_wmma_f32_16x16x4_f32(false, a, false, b, (short)0, c,
                                               false, false);
}

__device__ inline float fast_sigmoid(float p) {
  return __builtin_amdgcn_rcpf(1.f + __expf(-p));
}
__device__ inline float silu_f(float p) { return p * fast_sigmoid(p); }

__device__ inline void ace_sph(float x, float y, float z, float* sh) {
  float x2 = x * x, y2 = y * y, z2 = z * z;
  sh[0]  = 0.28209479f;
  sh[1]  = 0.48860251f * y;
  sh[2]  = 0.48860251f * z;
  sh[3]  = 0.48860251f * x;
  sh[4]  = 1.09254843f * x * y;
  sh[5]  = 1.09254843f * y * z;
  sh[6]  = 0.31539157f * (3.f * z2 - 1.f);
  sh[7]  = 1.09254843f * x * z;
  sh[8]  = 0.54627422f * (x2 - y2);
  sh[9]  = 0.59004359f * y * (3.f * x2 - y2);
  sh[10] = 2.89061144f * x * y * z;
  sh[11] = 0.45704579f * y * (5.f * z2 - 1.f);
  sh[12] = 0.37317633f * z * (5.f * z2 - 3.f);
  sh[13] = 0.45704579f * x * (5.f * z2 - 1.f);
  sh[14] = 1.44530572f * z * (x2 - y2);
  sh[15] = 0.59004359f * x * (x2 - 3.f * y2);
}

// geometry: vec, r (via v_rsq), unit vec, cutoff poly, sin/cos(n*theta) by
// angle-addition recurrence (one native sin + cos total)
__device__ inline void ace_geom(const float* __restrict__ pos,
                                const float* __restrict__ shifts, int e, int s,
                                int t, float& r, float& invr, float& ux,
                                float& uy, float& uz, float& cut, float& cutp,
                                float* sinv, float* cosv) {
  float vx = pos[3 * t + 0] - pos[3 * s + 0] + shifts[3 * e + 0];
  float vy = pos[3 * t + 1] - pos[3 * s + 1] + shifts[3 * e + 1];
  float vz = pos[3 * t + 2] - pos[3 * s + 2] + shifts[3 * e + 2];
  float r2 = vx * vx + vy * vy + vz * vz + 1e-18f;
  invr = __builtin_amdgcn_rsqf(r2);
  r = r2 * invr;
  ux = vx * invr; uy = vy * invr; uz = vz * invr;
  float u = r * (1.f / RMAX);
  cut = 0.f; cutp = 0.f;
  if (u < 1.f) {
    float u2 = u * u, u5 = u2 * u2 * u, u6 = u5 * u;
    cut = 1.f - 28.f * u6 + 48.f * u6 * u - 21.f * u6 * u2;
    cutp = -168.f * u5 + 336.f * u6 - 168.f * u6 * u;
  }
  float th = PI_F * r * (1.f / RMAX);
  float s1 = __sinf(th), c1 = __cosf(th);
  sinv[0] = s1; cosv[0] = c1;
#pragma unroll
  for (int n = 1; n < NB; ++n) {
    sinv[n] = sinv[n - 1] * c1 + cosv[n - 1] * s1;
    cosv[n] = cosv[n - 1] * c1 - sinv[n - 1] * s1;
  }
}

// ---------------- prep: symmetrized tensors + collapsed readout vectors -----
__global__ void ace_prep(const float* __restrict__ U2, const float* __restrict__ U3,
                         const float* __restrict__ Wp1, const float* __restrict__ Wp2,
                         const float* __restrict__ Wp3, const float* __restrict__ Wread,
                         float* __restrict__ Vmat, float* __restrict__ U2s,
                         float* __restrict__ wv) {
  int tid = threadIdx.x;
  for (int idx = tid; idx < 4096; idx += blockDim.x) {
    int i = idx & 15, pq = idx >> 4, p = pq >> 4, q = pq & 15;
    Vmat[pq * 16 + i] = U3[(i * 16 + p) * 16 + q] + U3[(p * 16 + i) * 16 + q] +
                        U3[(p * 16 + q) * 16 + i];
  }
  for (int idx = tid; idx < 256; idx += blockDim.x) {
    int i = idx >> 4, j = idx & 15;
    U2s[idx] = U2[i * 16 + j] + U2[j * 16 + i];
  }
  if (tid < 96) {
    int which = tid >> 5, k = tid & 31;
    const float* Wp = (which == 0) ? Wp1 : ((which == 1) ? Wp2 : Wp3);
    float acc = 0.f;
    for (int m = 0; m < 32; ++m) acc += Wp[k * 32 + m] * Wread[m];
    wv[tid] = acc;  // w1 | w2 | w3
  }
}

// ---------------- node embedding (WMMA): y = x@W_up, e0 + x.Wread -----------
__global__ __launch_bounds__(128) void ace_node_embed(
    const float* __restrict__ attrs, const float* __restrict__ ae,
    const float* __restrict__ W_embed, const float* __restrict__ W_up,
    const float* __restrict__ Wread, const int* __restrict__ batch,
    float* __restrict__ y, float* __restrict__ e_out) {
  __shared__ float sWu[KC * KC];
  __shared__ float sWe[NELEM * KC];
  __shared__ float sAe[NELEM];
  __shared__ float sWr[KC];
  int tid = threadIdx.x;
  for (int i = tid; i < KC * KC; i += 128) sWu[i] = W_up[i];
  for (int i = tid; i < NELEM * KC; i += 128) sWe[i] = W_embed[i];
  if (tid < NELEM) sAe[tid] = ae[tid];
  if (tid < KC) sWr[tid] = Wread[tid];
  __syncthreads();
  int wave = tid >> 5, lane = tid & 31;
  int row = lane & 15, col = row;
  bool hi = lane >= 16;
  int mbase = hi ? 8 : 0;
  int khalf = hi ? 2 : 0;
  int nbase = blockIdx.x * 64 + wave * 16;
  int n = nbase + row;
  float x[KC];
#pragma unroll
  for (int k = 0; k < KC; ++k) x[k] = 0.f;
  float e = 0.f;
#pragma unroll
  for (int c = 0; c < NELEM; ++c) {
    float a = attrs[(size_t)n * NELEM + c];
    e += a * sAe[c];
#pragma unroll
    for (int k = 0; k < KC; ++k) x[k] += a * sWe[c * KC + k];
  }
#pragma unroll
  for (int k = 0; k < KC; ++k) e += x[k] * sWr[k];
  // y = x @ W_up   ([16x32]@[32x32], 2 N-tiles x 8 K-chunks)
  v8f y0 = {}, y1 = {};
#pragma unroll
  for (int c4 = 0; c4 < 8; ++c4) {
    int kk = 4 * c4 + khalf;
    v2f a, b;
    a[0] = hi ? x[4 * c4 + 2] : x[4 * c4 + 0];
    a[1] = hi ? x[4 * c4 + 3] : x[4 * c4 + 1];
    b[0] = sWu[kk * KC + col];      b[1] = sWu[(kk + 1) * KC + col];      y0 = wmma4(a, b, y0);
    b[0] = sWu[kk * KC + col + 16]; b[1] = sWu[(kk + 1) * KC + col + 16]; y1 = wmma4(a, b, y1);
  }
#pragma unroll
  for (int v = 0; v < 8; ++v) {
    int m = mbase + v;
    size_t yb = (size_t)(nbase + m) * KC;
    y[yb + col] = y0[v];
    y[yb + col + 16] = y1[v];
  }
  if (!hi) atomicAdd(&e_out[batch[n]], e);
}

// ---------------- edge forward (WMMA radial MLP): scatter into A ------------
__global__ __launch_bounds__(128) void ace_edge_fwd(
    const float* __restrict__ pos, const int* __restrict__ ei,
    const float* __restrict__ shifts, const float* __restrict__ y,
    const float* __restrict__ Wr1, const float* __restrict__ br1,
    const float* __restrict__ Wr2, float* __restrict__ A) {
  __shared__ float sW1[NB * 64];
  __shared__ float sb1[64];
  __shared__ float sW2[64 * KC];
  __shared__ float Hs[4][16][64];   // silu(H)
  __shared__ float Ss[4][16][16];   // sh per edge
  __shared__ int Esnd[4][16];
  __shared__ int Ercv[4][16];
  int tid = threadIdx.x;
  for (int i = tid; i < NB * 64; i += 128) sW1[i] = Wr1[i];
  if (tid < 64) sb1[tid] = br1[tid];
  for (int i = tid; i < 64 * KC; i += 128) sW2[i] = Wr2[i];
  __syncthreads();

  int wave = tid >> 5, lane = tid & 31;
  int row = lane & 15, col = row;
  bool hi = lane >= 16;
  int khalf = hi ? 2 : 0;
  int mbase = hi ? 8 : 0;
  int e = blockIdx.x * 64 + wave * 16 + row;
  int s = ei[e], t = ei[NE + e];
  float r, invr, ux, uy, uz, cut, cutp, sinv[NB], cosv[NB];
  ace_geom(pos, shifts, e, s, t, r, invr, ux, uy, uz, cut, cutp, sinv, cosv);
  float Rb[NB];
#pragma unroll
  for (int n = 0; n < NB; ++n) Rb[n] = SQ2R * sinv[n] * invr * cut;
  float sh[NSHC];
  ace_sph(ux, uy, uz, sh);
  if (!hi) {
    Esnd[wave][row] = s;
    Ercv[wave][row] = t;
#pragma unroll
    for (int i = 0; i < NSHC; ++i) Ss[wave][row][i] = sh[i];
  }

  // layer 1: Hpre = Rb @ W1  ([16x8]@[8x64], 4 N-tiles x 2 K-chunks)
  v8f h0 = {}, h1 = {}, h2 = {}, h3 = {};
#pragma unroll
  for (int c4 = 0; c4 < 2; ++c4) {
    int kk = 4 * c4 + khalf;
    v2f a, b;
    a[0] = hi ? Rb[4 * c4 + 2] : Rb[4 * c4 + 0];
    a[1] = hi ? Rb[4 * c4 + 3] : Rb[4 * c4 + 1];
    b[0] = sW1[kk * 64 + col +  0]; b[1] = sW1[(kk + 1) * 64 + col +  0]; h0 = wmma4(a, b, h0);
    b[0] = sW1[kk * 64 + col + 16]; b[1] = sW1[(kk + 1) * 64 + col + 16]; h1 = wmma4(a, b, h1);
    b[0] = sW1[kk * 64 + col + 32]; b[1] = sW1[(kk + 1) * 64 + col + 32]; h2 = wmma4(a, b, h2);
    b[0] = sW1[kk * 64 + col + 48]; b[1] = sW1[(kk + 1) * 64 + col + 48]; h3 = wmma4(a, b, h3);
  }
#pragma unroll
  for (int v = 0; v < 8; ++v) {
    int m = mbase + v;
    Hs[wave][m][col +  0] = silu_f(h0[v] + sb1[col +  0]);
    Hs[wave][m][col + 16] = silu_f(h1[v] + sb1[col + 16]);
    Hs[wave][m][col + 32] = silu_f(h2[v] + sb1[col + 32]);
    Hs[wave][m][col + 48] = silu_f(h3[v] + sb1[col + 48]);
  }
  // layer 2: Rw = H @ W2  ([16x64]@[64x32], 2 N-tiles x 16 K-chunks)
  v8f d0 = {}, d1 = {};
#pragma unroll
  for (int c4 = 0; c4 < 16; ++c4) {
    int kk = 4 * c4 + khalf;
    v2f a, b;
    a[0] = Hs[wave][row][kk];
    a[1] = Hs[wave][row][kk + 1];
    b[0] = sW2[kk * KC + col];      b[1] = sW2[(kk + 1) * KC + col];      d0 = wmma4(a, b, d0);
    b[0] = sW2[kk * KC + col + 16]; b[1] = sW2[(kk + 1) * KC + col + 16]; d1 = wmma4(a, b, d1);
  }
  // scatter: A[t,k,i] += y[s,k]*Rw[k]*sh[i]/16
#pragma unroll
  for (int v = 0; v < 8; ++v) {
    int m = mbase + v;
    int sm = Esnd[wave][m], tm = Ercv[wave][m];
    {
      int k = col;
      float xs = y[(size_t)sm * KC + k] * d0[v] * INV_AVG;
      float* Ab = A + ((size_t)tm * KC + k) * NSHC;
#pragma unroll
      for (int i = 0; i < NSHC; ++i) atomicAdd(&Ab[i], xs * Ss[wave][m][i]);
    }
    {
      int k = col + 16;
      float xs = y[(size_t)sm * KC + k] * d1[v] * INV_AVG;
      float* Ab = A + ((size_t)tm * KC + k) * NSHC;
#pragma unroll
      for (int i = 0; i < NSHC; ++i) atomicAdd(&Ab[i], xs * Ss[wave][m][i]);
    }
  }
}

// ---------------- WMMA node einsum: energy B-terms + gA ---------------------
__global__ __launch_bounds__(128) void ace_node_einsum(
    const float* __restrict__ A, const float* __restrict__ U2,
    const float* __restrict__ Vmat, const float* __restrict__ U2s,
    const float* __restrict__ wv, const int* __restrict__ batch,
    float* __restrict__ gA, float* __restrict__ e_out) {
  __shared__ float Vs[4096];
  __shared__ float U2l[256];
  __shared__ float U2sl[256];
  __shared__ float wls[96];
  __shared__ float Xs[4][16][16];
  int tid = threadIdx.x;
  for (int i = tid; i < 4096; i += 128) Vs[i] = Vmat[i];
  for (int i = tid; i < 256; i += 128) {
    U2l[i] = U2[i];
    U2sl[i] = U2s[i];
  }
  if (tid < 96) wls[tid] = wv[tid];
  __syncthreads();

  int wave = tid >> 5, lane = tid & 31;
  int tile = blockIdx.x * 4 + wave;
  int n = tile >> 1;
  int kbase = (tile & 1) << 4;
  const float* Arow = A + ((size_t)n * KC + kbase) * NSHC;
  for (int i = lane; i < 256; i += 32) Xs[wave][i >> 4][i & 15] = Arow[i];
  __syncthreads();

  int row = lane & 15;
  bool hi = lane >= 16;
  int col = row;
  float xr[16];
#pragma unroll
  for (int i = 0; i < 16; ++i) xr[i] = Xs[wave][row][i];

  v8f c3 = {};  // gA3 = O @ Vmat  (K = 256)
  v8f c2 = {};  // P2  = X @ U2    (K = 16)
  v8f cg = {};  // G2  = X @ U2s^T (K = 16)
#pragma unroll
  for (int c4 = 0; c4 < 64; ++c4) {
    int t0 = 4 * c4;
    float o0 = xr[(t0 + 0) >> 4] * xr[(t0 + 0) & 15];
    float o1 = xr[(t0 + 1) >> 4] * xr[(t0 + 1) & 15];
    float o2 = xr[(t0 + 2) >> 4] * xr[(t0 + 2) & 15];
    float o3 = xr[(t0 + 3) >> 4] * xr[(t0 + 3) & 15];
    v2f a, b;
    a[0] = hi ? o2 : o0;
    a[1] = hi ? o3 : o1;
    int kk = t0 + (hi ? 2 : 0);
    b[0] = Vs[kk * 16 + col];
    b[1] = Vs[(kk + 1) * 16 + col];
    c3 = wmma4(a, b, c3);
  }
#pragma unroll
  for (int c4 = 0; c4 < 4; ++c4) {
    int k0 = 4 * c4;
    int kk = k0 + (hi ? 2 : 0);
    v2f a, b2v, bgv;
    a[0] = hi ? xr[k0 + 2] : xr[k0 + 0];
    a[1] = hi ? xr[k0 + 3] : xr[k0 + 1];
    b2v[0] = U2l[kk * 16 + col];
    b2v[1] = U2l[(kk + 1) * 16 + col];
    c2 = wmma4(a, b2v, c2);
    bgv[0] = U2sl[col * 16 + kk];
    bgv[1] = U2sl[col * 16 + kk + 1];
    cg = wmma4(a, bgv, cg);
  }

  float epart = 0.f;
  int mhalf = hi ? 8 : 0;
#pragma unroll
  for (int v = 0; v < 8; ++v) {
    int m = v + mhalf;
    float xm = Xs[wave][m][col];
    float b3p = c3[v] * xm;
    float b2p = c2[v] * xm;
#pragma unroll
    for (int off = 1; off < 16; off <<= 1) {
      b3p += __shfl_xor(b3p, off);
      b2p += __shfl_xor(b2p, off);
    }
    int k = kbase + m;
    float w1 = wls[k], w2 = wls[32 + k], w3 = wls[64 + k];
    if (col == 0)
      epart += w1 * Xs[wave][m][0] + w2 * b2p + w3 * (b3p * (1.f / 3.f));
    float gval = w3 * c3[v] + w2 * cg[v] + ((col == 0) ? w1 : 0.f);
    gA[((size_t)n * KC + k) * NSHC + col] = gval;
  }
  epart += __shfl_xor(epart, 16);
  if (lane == 0) atomicAdd(&e_out[batch[n]], epart);
}

// ---------------- edge backward (WMMA radial MLP bwd): forces ---------------
__global__ __launch_bounds__(128) void ace_edge_bwd(
    const float* __restrict__ pos, const int* __restrict__ ei,
    const float* __restrict__ shifts, const float* __restrict__ y,
    const float* __restrict__ Wr1, const float* __restrict__ br1,
    const float* __restrict__ Wr2, const float* __restrict__ gA,
    float* __restrict__ forces) {
  __shared__ float sW1[NB * 64];
  __shared__ float sb1[64];
  __shared__ float sW2[64 * KC];
  __shared__ float Hs[4][16][64];   // silu(H) fwd; reused for gpre in bwd
  __shared__ float Rws[4][16][32];  // Rw per edge
  __shared__ float Gs[4][16][32];   // gRw per edge
  __shared__ float GRs[4][16][8];   // gR per edge
  int tid = threadIdx.x;
  for (int i = tid; i < NB * 64; i += 128) sW1[i] = Wr1[i];
  if (tid < 64) sb1[tid] = br1[tid];
  for (int i = tid; i < 64 * KC; i += 128) sW2[i] = Wr2[i];
  __syncthreads();

  int wave = tid >> 5, lane = tid & 31;
  int row = lane & 15, col = row;
  bool hi = lane >= 16;
  int khalf = hi ? 2 : 0;
  int mbase = hi ? 8 : 0;
  int e = blockIdx.x * 64 + wave * 16 + row;
  int s = ei[e], t = ei[NE + e];
  __builtin_prefetch(gA + (size_t)t * KC * NSHC, 0, 0);
  float r, invr, ux, uy, uz, cut, cutp, sinv[NB], cosv[NB];
  ace_geom(pos, shifts, e, s, t, r, invr, ux, uy, uz, cut, cutp, sinv, cosv);
  float Rb[NB];
#pragma unroll
  for (int n = 0; n < NB; ++n) Rb[n] = SQ2R * sinv[n] * invr * cut;
  float sh[NSHC];
  ace_sph(ux, uy, uz, sh);

  // forward layer 1 (keep pre-activations in C-layout registers)
  v8f h0 = {}, h1 = {}, h2 = {}, h3 = {};
#pragma unroll
  for (int c4 = 0; c4 < 2; ++c4) {
    int kk = 4 * c4 + khalf;
    v2f a, b;
    a[0] = hi ? Rb[4 * c4 + 2] : Rb[4 * c4 + 0];
    a[1] = hi ? Rb[4 * c4 + 3] : Rb[4 * c4 + 1];
    b[0] = sW1[kk * 64 + col +  0]; b[1] = sW1[(kk + 1) * 64 + col +  0]; h0 = wmma4(a, b, h0);
    b[0] = sW1[kk * 64 + col + 16]; b[1] = sW1[(kk + 1) * 64 + col + 16]; h1 = wmma4(a, b, h1);
    b[0] = sW1[kk * 64 + col + 32]; b[1] = sW1[(kk + 1) * 64 + col + 32]; h2 = wmma4(a, b, h2);
    b[0] = sW1[kk * 64 + col + 48]; b[1] = sW1[(kk + 1) * 64 + col + 48]; h3 = wmma4(a, b, h3);
  }
#pragma unroll
  for (int v = 0; v < 8; ++v) {
    int m = mbase + v;
    float p;
    p = h0[v] + sb1[col +  0]; h0[v] = p; Hs[wave][m][col +  0] = silu_f(p);
    p = h1[v] + sb1[col + 16]; h1[v] = p; Hs[wave][m][col + 16] = silu_f(p);
    p = h2[v] + sb1[col + 32]; h2[v] = p; Hs[wave][m][col + 32] = silu_f(p);
    p = h3[v] + sb1[col + 48]; h3[v] = p; Hs[wave][m][col + 48] = silu_f(p);
  }
  // forward layer 2: Rw, stash per-edge rows to LDS
  v8f d0 = {}, d1 = {};
#pragma unroll
  for (int c4 = 0; c4 < 16; ++c4) {
    int kk = 4 * c4 + khalf;
    v2f a, b;
    a[0] = Hs[wave][row][kk];
    a[1] = Hs[wave][row][kk + 1];
    b[0] = sW2[kk * KC + col];      b[1] = sW2[(kk + 1) * KC + col];      d0 = wmma4(a, b, d0);
    b[0] = sW2[kk * KC + col + 16]; b[1] = sW2[(kk + 1) * KC + col + 16]; d1 = wmma4(a, b, d1);
  }
#pragma unroll
  for (int v = 0; v < 8; ++v) {
    int m = mbase + v;
    Rws[wave][m][col] = d0[v];
    Rws[wave][m][col + 16] = d1[v];
  }

  // gather gA: halves split the 32 channels of this lane's own edge
  // (fully unrolled so all register arrays stay constant-indexed)
  int kb = hi ? 16 : 0;
  float gsh[NSHC];
#pragma unroll
  for (int i = 0; i < NSHC; ++i) gsh[i] = 0.f;
  const float* gAr = gA + (size_t)t * KC * NSHC;
#pragma unroll
  for (int kk2 = 0; kk2 < 16; ++kk2) {
    int k = kb + kk2;
    float yv = y[(size_t)s * KC + k];
    float xsk = yv * Rws[wave][row][k];
    const float* g = gAr + k * NSHC;
    float acc = 0.f;
#pragma unroll
    for (int i = 0; i < NSHC; ++i) {
      float gv = g[i] * INV_AVG;
      acc += gv * sh[i];
      gsh[i] += gv * xsk;
    }
    Gs[wave][row][k] = acc * yv;  // gRw[k] = gxs[k]*y[s,k]
  }
#pragma unroll
  for (int i = 0; i < NSHC; ++i) gsh[i] += __shfl_xor(gsh[i], 16);

  // gH = gRw @ W2^T  ([16x32]@[32x64], 4 N-tiles x 8 K-chunks)
  v8f q0 = {}, q1 = {}, q2 = {}, q3 = {};
#pragma unroll
  for (int c4 = 0; c4 < 8; ++c4) {
    int kk = 4 * c4 + khalf;
    v2f a, b;
    a[0] = Gs[wave][row][kk];
    a[1] = Gs[wave][row][kk + 1];
    b[0] = sW2[(col +  0) * KC + kk]; b[1] = sW2[(col +  0) * KC + kk + 1]; q0 = wmma4(a, b, q0);
    b[0] = sW2[(col + 16) * KC + kk]; b[1] = sW2[(col + 16) * KC + kk + 1]; q1 = wmma4(a, b, q1);
    b[0] = sW2[(col + 32) * KC + kk]; b[1] = sW2[(col + 32) * KC + kk + 1]; q2 = wmma4(a, b, q2);
    b[0] = sW2[(col + 48) * KC + kk]; b[1] = sW2[(col + 48) * KC + kk + 1]; q3 = wmma4(a, b, q3);
  }
  // gpre = gH * silu'(pre); tiles align elementwise; write into Hs (reuse)
#pragma unroll
  for (int v = 0; v < 8; ++v) {
    int m = mbase + v;
    float p, sg, ds;
    p = h0[v]; sg = fast_sigmoid(p); ds = sg * (1.f + p * (1.f - sg));
    Hs[wave][m][col +  0] = q0[v] * ds;
    p = h1[v]; sg = fast_sigmoid(p); ds = sg * (1.f + p * (1.f - sg));
    Hs[wave][m][col + 16] = q1[v] * ds;
    p = h2[v]; sg = fast_sigmoid(p); ds = sg * (1.f + p * (1.f - sg));
    Hs[wave][m][col + 32] = q2[v] * ds;
    p = h3[v]; sg = fast_sigmoid(p); ds = sg * (1.f + p * (1.f - sg));
    Hs[wave][m][col + 48] = q3[v] * ds;
  }
  // gR = gpre @ W1^T  ([16x64]@[64x8], N padded to 16, 16 K-chunks)
  v8f qr = {};
#pragma unroll
  for (int c4 = 0; c4 < 16; ++c4) {
    int kk = 4 * c4 + khalf;
    v2f a, b;
    a[0] = Hs[wave][row][kk];
    a[1] = Hs[wave][row][kk + 1];
    b[0] = (col < 8) ? sW1[col * 64 + kk] : 0.f;
    b[1] = (col < 8) ? sW1[col * 64 + kk + 1] : 0.f;
    qr = wmma4(a, b, qr);
  }
#pragma unroll
  for (int v = 0; v < 8; ++v) {
    int m = mbase + v;
    if (col < 8) GRs[wave][m][col] = qr[v];
  }

  // per-edge finalization (both halves compute; low half writes atomics)
  float gR[NB];
#pragma unroll
  for (int n = 0; n < NB; ++n) gR[n] = GRs[wave][row][n];
  float g_r = 0.f;
#pragma unroll
  for (int n = 0; n < NB; ++n) {
    float w = PI_F * (float)(n + 1) * (1.f / RMAX);
    float sn = sinv[n], cn = cosv[n];
    float db = SQ2R * (w * cn * r - sn) * invr * invr;
    float dR = db * cut + SQ2R * sn * invr * cutp * (1.f / RMAX);
    g_r += gR[n] * dR;
  }
  float x2 = ux * ux, y2 = uy * uy, z2 = uz * uz;
  float gx = 0.48860251f * gsh[3] + 1.09254843f * (uy * gsh[4] + uz * gsh[7]) +
             1.09254844f * ux * gsh[8] + 3.54026154f * ux * uy * gsh[9] +
             2.89061144f * uy * uz * gsh[10] +
             0.45704579f * (5.f * z2 - 1.f) * gsh[13] +
             2.89061144f * ux * uz * gsh[14] +
             1.77013077f * (x2 - y2) * gsh[15];
  float gy = 0.48860251f * gsh[1] + 1.09254843f * (ux * gsh[4] + uz * gsh[5]) -
             1.09254844f * uy * gsh[8] + 1.77013077f * (x2 - y2) * gsh[9] +
             2.89061144f * ux * uz * gsh[10] +
             0.45704579f * (5.f * z2 - 1.f) * gsh[11] -
             2.89061144f * uy * uz * gsh[14] - 3.54026154f * ux * uy * gsh[15];
  float gz = 0.48860251f * gsh[2] + 1.09254843f * (uy * gsh[5] + ux * gsh[7]) +
             1.89234942f * uz * gsh[6] + 2.89061144f * ux * uy * gsh[10] +
             4.5704579f * uy * uz * gsh[11] +
             0.37317633f * (15.f * z2 - 3.f) * gsh[12] +
             4.5704579f * ux * uz * gsh[13] + 1.44530572f * (x2 - y2) * gsh[14];
  float dotgu = gx * ux + gy * uy + gz * uz;
  float gvx = g_r * ux + (gx - dotgu * ux) * invr;
  float gvy = g_r * uy + (gy - dotgu * uy) * invr;
  float gvz = g_r * uz + (gz - dotgu * uz) * invr;
  if (!hi) {
    atomicAdd(&forces[3 * t + 0], -gvx);
    atomicAdd(&forces[3 * t + 1], -gvy);
    atomicAdd(&forces[3 * t + 2], -gvz);
    atomicAdd(&forces[3 * s + 0], gvx);
    atomicAdd(&forces[3 * s + 1], gvy);
    atomicAdd(&forces[3 * s + 2], gvz);
  }
}

extern "C" void kernel_launch(void* const* d_in, const int* in_sizes, int n_in,
                              void* d_out, int out_size, void* d_ws, size_t ws_size,
                              hipStream_t stream) {
  (void)in_sizes; (void)n_in; (void)out_size; (void)ws_size;
  const float* pos    = (const float*)d_in[0];
  const float* attrs  = (const float*)d_in[1];
  const int*   ei     = (const int*)d_in[2];
  const float* shifts = (const float*)d_in[3];
  const int*   batch  = (const int*)d_in[4];
  const float* ae     = (const float*)d_in[6];
  const float* W_embed= (const float*)d_in[7];
  const float* W_up   = (const float*)d_in[8];
  const float* Wr1    = (const float*)d_in[9];
  const float* br1    = (const float*)d_in[10];
  const float* Wr2    = (const float*)d_in[11];
  const float* U2     = (const float*)d_in[12];
  const float* U3     = (const float*)d_in[13];
  const float* Wp1    = (const float*)d_in[14];
  const float* Wp2    = (const float*)d_in[15];
  const float* Wp3    = (const float*)d_in[16];
  const float* Wread  = (const float*)d_in[17];

  float* out    = (float*)d_out;
  float* e_out  = out;        // [16]
  float* forces = out + NGR;  // [NN*3]

  float* ws   = (float*)d_ws;
  float* A    = ws + OFF_A;
  float* gA   = ws + OFF_GA;
  float* y    = ws + OFF_Y;
  float* Vmat = ws + OFF_VM;
  float* U2s  = ws + OFF_U2S;
  float* wv   = ws + OFF_WV;

  hipMemsetAsync(d_out, 0, (size_t)(NGR + (size_t)NN * 3) * sizeof(float), stream);
  hipMemsetAsync(A, 0, SZ_A * sizeof(float), stream);

  ace_prep<<<1, 256, 0, stream>>>(U2, U3, Wp1, Wp2, Wp3, Wread, Vmat, U2s, wv);
  ace_node_embed<<<NN / 64, 128, 0, stream>>>(attrs, ae, W_embed, W_up, Wread,
                                              batch, y, e_out);
  ace_edge_fwd<<<NE / 64, 128, 0, stream>>>(pos, ei, shifts, y, Wr1, br1, Wr2, A);
  ace_node_einsum<<<(NN * 2) / 4, 128, 0, stream>>>(A, U2, Vmat, U2s, wv, batch,
                                                    gA, e_out);
  ace_edge_bwd<<<NE / 64, 128, 0, stream>>>(pos, ei, shifts, y, Wr1, br1, Wr2,
                                            gA, forces);
}